// WaveletTransform_4217657884944
// MI455X (gfx1250) — compile-verified
//
#include <hip/hip_runtime.h>

// Level-1 Haar DWT, (32, 4096, 512) f32 -> (32, 4096, 512) f32.
// out[b, p,      f] = (x[b,2p,f] + x[b,2p+1,f]) * invsqrt2   (p in [0,2048))
// out[b, 2048+p, f] = (x[b,2p,f] - x[b,2p+1,f]) * invsqrt2
//
// Formulated as D(16x16) = W(16x16) * X(16x16) per tile, computed with
// 4 chained V_WMMA_F32_16X16X4_F32 (K=4 chunks). W rows 0-7 = cA combos,
// rows 8-15 = cD combos, entries +-1/sqrt(2). Memory traffic identical to
// the elementwise formulation (each element read once, written once).

typedef float v2f __attribute__((ext_vector_type(2)));
typedef float v8f __attribute__((ext_vector_type(8)));

#define FDIM   512
#define LDIM   4096
#define HALF_L 2048

// W[m][k]: Haar butterfly matrix for a 16-row L block.
// m<8 : row m has +s at k=2m and +s at k=2m+1        (cA)
// m>=8: row m has +s at k=2(m-8) and -s at k=2(m-8)+1 (cD)
__device__ __forceinline__ float haarW(int m, int k) {
  const float s = 0.70710678118654752440f;  // float32(1/sqrt(2))
  return ((k >> 1) == (m & 7)) ? (((m >= 8) && (k & 1)) ? -s : s) : 0.0f;
}

__global__ __launch_bounds__(256) void haar_dwt_wmma(const float* __restrict__ x,
                                                     float* __restrict__ out) {
  const int lane = threadIdx.x & 31;
  const int wave = threadIdx.x >> 5;
  const int half = lane >> 4;   // which 16-lane half of the wave
  const int mn   = lane & 15;   // row m (A / D) == column n (B / D)

  // One 16(L) x 16(F) tile per wave. 32 batches * 256 L-tiles * 32 F-tiles.
  const int tile = blockIdx.x * 8 + wave;
  const int fblk = tile & 31;          // 512/16  = 32
  const int lblk = (tile >> 5) & 255;  // 4096/16 = 256
  const int b    = tile >> 13;         // 32 batches (256*32 = 2^13 tiles/batch)

  const int f0 = fblk << 4;
  const int L0 = lblk << 4;
  const int P0 = lblk << 3;  // output pair-row base = L0/2

  // B-operand base: column n = mn, rows k added per chunk.
  const float* xbase = x + ((size_t)b * LDIM + L0) * FDIM + f0 + mn;

  v8f c = {};  // f32 16x16 accumulator (8 VGPRs)

#pragma unroll
  for (int kb = 0; kb < 16; kb += 4) {
    const int k0 = kb + 2 * half;  // K striping: lanes 0-15 hold k0,k0+1; lanes 16-31 hold k0+2,k0+3
    v2f a, bm;
    a.x = haarW(mn, k0);
    a.y = haarW(mn, k0 + 1);
    const float* px = xbase + (size_t)k0 * FDIM;
    bm.x = __builtin_nontemporal_load(px);          // X[k0  ][n]
    bm.y = __builtin_nontemporal_load(px + FDIM);   // X[k0+1][n]
    // D = A*B + C ; args: (neg_a, A, neg_b, B, c_mod, C, reuse_a, reuse_b)
    c = __builtin_amdgcn_wmma_f32_16x16x4_f32(false, a, false, bm, (short)0, c,
                                              false, false);
  }

  // D layout: VGPR j -> M=j (lanes 0-15), M=j+8 (lanes 16-31); N = mn.
  // M<8 => cA row P0+M ; M>=8 => cD row 2048+P0+(M-8).
  const size_t outRow0 = (size_t)b * LDIM + P0 + (half ? HALF_L : 0);
  float* pout = out + outRow0 * FDIM + f0 + mn;
#pragma unroll
  for (int j = 0; j < 8; ++j) {
    __builtin_nontemporal_store(c[j], pout + (size_t)j * FDIM);
  }
}

extern "C" void kernel_launch(void* const* d_in, const int* in_sizes, int n_in,
                              void* d_out, int out_size, void* d_ws, size_t ws_size,
                              hipStream_t stream) {
  const float* x = (const float*)d_in[0];
  float* out = (float*)d_out;
  // 262144 tiles total, 8 waves (= 8 tiles) per 256-thread block.
  dim3 block(256);
  dim3 grid(262144 / 8);
  haar_dwt_wmma<<<grid, block, 0, stream>>>(x, out);
}